// ce_loss_aux_72499047956668
// MI455X (gfx1250) — compile-verified
//
#include <hip/hip_runtime.h>

typedef __attribute__((ext_vector_type(2))) float v2f;
typedef __attribute__((ext_vector_type(4))) float v4f;
typedef __attribute__((ext_vector_type(8))) float v8f;

#define THREADS 256
#define POS_PER_BLOCK 4096            // positions per block (divides L=8192)
#define F4_PER_THREAD 8               // POS_PER_BLOCK / (2 positions-per-float4 * THREADS)

// Stage 1: per-block partial sum of log2(selected prob) over valid positions.
// Grid is 2D: blockIdx.y = row b (doc_len uniform -> scalar load),
//             blockIdx.x = segment within the row. No integer division anywhere.
template <bool GUARD>
__global__ __launch_bounds__(THREADS)
void ce_partial_kernel(const float* __restrict__ y_true,
                       const float* __restrict__ y_pred,
                       const int*   __restrict__ doc_len,
                       int L,
                       float* __restrict__ partials)
{
    const int tid  = threadIdx.x;
    const int b    = blockIdx.y;
    const int seg  = blockIdx.x;
    const int dlen = doc_len[b];

    // float4 indexing: each float4 holds 2 positions (2 floats per position)
    const long long f4base =
        (long long)b * (L >> 1) + (long long)seg * (POS_PER_BLOCK / 2) + tid;
    const int relBase = seg * POS_PER_BLOCK + tid * 2;

    const v4f* __restrict__ yp4 = (const v4f*)y_pred;
    const v4f* __restrict__ yt4 = (const v4f*)y_true;

    float s = 0.0f;   // sum of log2(selected p) over valid positions
#pragma unroll
    for (int i = 0; i < F4_PER_THREAD; ++i) {
        const long long f4 = f4base + (long long)i * THREADS;
        const int rel0 = relBase + i * (2 * THREADS);
        if (!GUARD || rel0 < L) {
            // single-use streaming data -> non-temporal 128-bit loads
            v4f p = __builtin_nontemporal_load(&yp4[f4]);
            v4f m = __builtin_nontemporal_load(&yt4[f4]);
            // one-hot: y_true[...,1]==1 -> class-1 prob, else class-0 prob
            float sel0 = (m.y == 1.0f) ? p.y : p.x;
            float sel1 = (m.w == 1.0f) ? p.w : p.z;
            // invalid positions contribute log2(1) == 0
            sel0 = (rel0     < dlen) ? sel0 : 1.0f;
            sel1 = (rel0 + 1 < dlen) ? sel1 : 1.0f;
            s += __log2f(sel0) + __log2f(sel1);
        }
    }

    // ---- 32-lane reduction on the matrix pipe: V_WMMA_F32_16X16X4_F32 ----
    // A (16x4 f32) layout: lane m (0-15) holds A[m][0..1]; lane m+16 holds A[m][2..3].
    // With A = (s, 0) per lane and B = all-ones (4x16):
    //   D[m][n] = sum_k A[m][k] = s[m] + s[m+16]   (identical for every column n)
    // Lanes 0-15 hold D rows 0..7 in their 8 regs, lanes 16-31 hold rows 8..15,
    // so (per-lane sum of D)[lane0] + (per-lane sum of D)[lane16] == sum of all 32 s.
    v2f a;  a.x = s;     a.y = 0.0f;
    v2f bb; bb.x = 1.0f; bb.y = 1.0f;
    v8f c = {};
    v8f d = __builtin_amdgcn_wmma_f32_16x16x4_f32(
        /*neg_a=*/false, a, /*neg_b=*/false, bb,
        /*c_mod=*/(short)0, c, /*reuse_a=*/false, /*reuse_b=*/false);
    float half8 = ((d[0] + d[1]) + (d[2] + d[3])) + ((d[4] + d[5]) + (d[6] + d[7]));

    __shared__ float red[16];               // 8 waves x 2 halves
    const int wave = tid >> 5;
    const int lane = tid & 31;
    if (lane == 0)  red[wave * 2 + 0] = half8;
    if (lane == 16) red[wave * 2 + 1] = half8;
    __syncthreads();
    if (tid == 0) {
        float t = 0.0f;
#pragma unroll
        for (int i = 0; i < 16; ++i) t += red[i];
        partials[blockIdx.y * gridDim.x + blockIdx.x] = t;
    }
}

// Stage 2: deterministic fold of block partials + sum(doc_len), final divide.
__global__ __launch_bounds__(THREADS)
void ce_finalize_kernel(const float* __restrict__ partials, int nPartials,
                        const int* __restrict__ doc_len, int nB,
                        float* __restrict__ out)
{
    __shared__ float sl[THREADS];
    __shared__ float st[THREADS];
    const int tid = threadIdx.x;

    float s = 0.0f;
    for (int i = tid; i < nPartials; i += THREADS) s += partials[i];
    float t = 0.0f;
    for (int i = tid; i < nB; i += THREADS) t += (float)doc_len[i];
    sl[tid] = s;
    st[tid] = t;
    __syncthreads();
#pragma unroll
    for (int off = THREADS / 2; off > 0; off >>= 1) {
        if (tid < off) { sl[tid] += sl[tid + off]; st[tid] += st[tid + off]; }
        __syncthreads();
    }
    if (tid == 0) {
        const float LN2 = 0.6931471805599453f;
        out[0] = (-LN2 * sl[0]) / st[0];
    }
}

extern "C" void kernel_launch(void* const* d_in, const int* in_sizes, int n_in,
                              void* d_out, int out_size, void* d_ws, size_t ws_size,
                              hipStream_t stream)
{
    const float* y_true  = (const float*)d_in[0];   // [B, L, 2] fp32 one-hot
    const float* y_pred  = (const float*)d_in[1];   // [B, L, 2] fp32 probs
    const int*   doc_len = (const int*)d_in[2];     // [B] int32

    const int nB = in_sizes[2];
    const long long ypElems = (long long)in_sizes[1];
    const int L = (int)(ypElems / (2LL * nB));      // 8192

    const int segs  = (L + POS_PER_BLOCK - 1) / POS_PER_BLOCK;
    const bool exact = (L % POS_PER_BLOCK) == 0;    // true for L=8192
    const int nPartials = segs * nB;

    float* partials = (float*)d_ws;                 // nPartials floats of scratch
    dim3 grid(segs, nB);

    if (exact) {
        ce_partial_kernel<false><<<grid, THREADS, 0, stream>>>(
            y_true, y_pred, doc_len, L, partials);
    } else {
        ce_partial_kernel<true><<<grid, THREADS, 0, stream>>>(
            y_true, y_pred, doc_len, L, partials);
    }
    ce_finalize_kernel<<<1, THREADS, 0, stream>>>(
        partials, nPartials, doc_len, nB, (float*)d_out);
}